// RelATE_52321291600091
// MI455X (gfx1250) — compile-verified
//
#include <hip/hip_runtime.h>
#include <hip/hip_bf16.h>
#include <math.h>

// Problem constants (from setup_inputs)
#define BB 4
#define NN 5
#define KK 5
#define QQ 5
#define NQC (NN*QQ)     // 25
#define LL 128
#define FF 768
#define KL (KK*LL)      // 640
#define EPSF 1e-8f
#define INV_BQ (1.0f/(BB*NQC))

typedef __attribute__((ext_vector_type(2))) float v2f;
typedef __attribute__((ext_vector_type(4))) float v4f;
typedef __attribute__((ext_vector_type(8))) float v8f;

// Workspace layout (float units)
#define OFF_ACC      0                            // [0]=sent_loss_acc [1]=token_loss_acc
#define OFF_PROTO    2                            // B*N*F sentence prototypes
#define OFF_TPROTO   (OFF_PROTO  + BB*NN*FF)      // B*NQ*4*F token prototypes
#define OFF_P2       (OFF_TPROTO + BB*NQC*4*FF)   // B*NQ*4
#define OFF_TLOGITS  (OFF_P2     + BB*NQC*4)      // B*NQ*L*4
#define OFF_END      (OFF_TLOGITS + BB*NQC*LL*4)  // ints (sent_pred) start here

// ---------------------------------------------------------------- k0: init
__global__ void k0_init(float* wsf) {
  if (threadIdx.x < 2) wsf[threadIdx.x] = 0.0f;
}

// ------------------------------------------- k1: sentence prototypes (B,N,F)
__global__ void k1_sent_proto(const float* __restrict__ sup, float* __restrict__ proto) {
  int bn = blockIdx.x;                       // b*N + n
  const float* base = sup + (size_t)bn * KK * LL * FF;   // [k][l=0][f]
  for (int f = threadIdx.x; f < FF; f += blockDim.x) {
    float s = 0.0f;
    for (int k = 0; k < KK; ++k) s += base[(size_t)k * LL * FF + f];
    proto[(size_t)bn * FF + f] = s * (1.0f / KK);
  }
}

// --------------------- k2: sentence logits, log-softmax loss, sent_pred
__global__ void k2_sent(const float* __restrict__ qemb, const float* __restrict__ proto,
                        const int* __restrict__ label, float* wsf, int* __restrict__ sent_pred) {
  int bq = blockIdx.x;
  int b  = bq / NQC;
  __shared__ float red[128];
  __shared__ float dv[NN];
  const float* qs = qemb + (size_t)bq * LL * FF;         // l = 0 row
  for (int n = 0; n < NN; ++n) {
    const float* pr = proto + (size_t)(b * NN + n) * FF;
    float part = 0.0f;
    for (int f = threadIdx.x; f < FF; f += 128) {
      float d = pr[f] - qs[f];
      part += d * d;
    }
    red[threadIdx.x] = part; __syncthreads();
    for (int s = 64; s > 0; s >>= 1) {
      if (threadIdx.x < s) red[threadIdx.x] += red[threadIdx.x + s];
      __syncthreads();
    }
    if (threadIdx.x == 0) dv[n] = red[0];
    __syncthreads();
  }
  if (threadIdx.x == 0) {
    float lg[NN], mx = -1e30f;
    for (int n = 0; n < NN; ++n) { lg[n] = -dv[n]; mx = fmaxf(mx, lg[n]); }
    float se = 0.0f;
    for (int n = 0; n < NN; ++n) se += __expf(lg[n] - mx);
    float lse = mx + __logf(se);
    int lab = label[bq];
    atomicAdd(&wsf[0], -(lg[lab] - lse) * INV_BQ);
    int best = 0; float bv = lg[0];
    for (int n = 1; n < NN; ++n) if (lg[n] > bv) { bv = lg[n]; best = n; }  // first max
    sent_pred[bq] = best;
  }
}

// ------------- k3: token prototypes (mask(4x640) . support(640xF)) + p2
__global__ void k3_token_proto(const float* __restrict__ sup, const int* __restrict__ s_ent,
                               const int* __restrict__ s_att, const int* __restrict__ label,
                               float* __restrict__ tproto, float* __restrict__ p2ws) {
  int bq = blockIdx.x;
  int b  = bq / NQC;
  int cond = label[bq];
  __shared__ float mask[4 * KL];   // 10 KB
  __shared__ float counts[4];
  __shared__ float red[256];
  const int* labp = s_ent + (size_t)(b * NN + cond) * KL;
  const int* attp = s_att + (size_t)(b * NN + cond) * KL;
  float cpart[4] = {0, 0, 0, 0};
  for (int kl = threadIdx.x; kl < KL; kl += 256) {
    int lv = labp[kl];
    int av = attp[kl];
    #pragma unroll
    for (int c = 0; c < 4; ++c) {
      float m = (lv == c + 1 && av > 0) ? 1.0f : 0.0f;
      mask[c * KL + kl] = m;
      cpart[c] += m;
    }
  }
  for (int c = 0; c < 4; ++c) {
    red[threadIdx.x] = cpart[c]; __syncthreads();
    for (int s = 128; s > 0; s >>= 1) {
      if (threadIdx.x < s) red[threadIdx.x] += red[threadIdx.x + s];
      __syncthreads();
    }
    if (threadIdx.x == 0) counts[c] = red[0];
    __syncthreads();
  }
  const float* supb = sup + (size_t)(b * NN + cond) * KL * FF;
  float p2part[4] = {0, 0, 0, 0};
  for (int f = threadIdx.x; f < FF; f += 256) {
    float acc[4] = {0, 0, 0, 0};
    for (int kl = 0; kl < KL; ++kl) {
      float s = supb[(size_t)kl * FF + f];   // coalesced across f-threads
      acc[0] += mask[0 * KL + kl] * s;
      acc[1] += mask[1 * KL + kl] * s;
      acc[2] += mask[2 * KL + kl] * s;
      acc[3] += mask[3 * KL + kl] * s;
    }
    #pragma unroll
    for (int c = 0; c < 4; ++c) {
      float v = acc[c] / (counts[c] + EPSF);
      tproto[((size_t)bq * 4 + c) * FF + f] = v;
      p2part[c] += v * v;
    }
  }
  for (int c = 0; c < 4; ++c) {
    __syncthreads();
    red[threadIdx.x] = p2part[c]; __syncthreads();
    for (int s = 128; s > 0; s >>= 1) {
      if (threadIdx.x < s) red[threadIdx.x] += red[threadIdx.x + s];
      __syncthreads();
    }
    if (threadIdx.x == 0) p2ws[bq * 4 + c] = red[0];
  }
}

// ------------- k4: qp einsum via V_WMMA_F32_16X16X4_F32, fused token_logits
// Per (b,q): A = query (128x768), B = proto^T (768x16, rows 4..15 zero-padded
// in LDS so every lane's B-fragment load is unconditional -> no EXEC toggling,
// single ds_load_b64 per fragment). 4 waves/block, 2 M-tiles/wave, K step 4.
__global__ void k4_qp_wmma(const float* __restrict__ qemb, const float* __restrict__ tproto,
                           const float* __restrict__ p2ws, float* __restrict__ tlog) {
  int bq = blockIdx.x;
  __shared__ float protoS[16 * FF];  // 48 KB, rows 4..15 zero
  __shared__ float q2S[LL];
  __shared__ float p2S[4];
  for (int i = threadIdx.x; i < 4 * FF; i += 128)
    protoS[i] = tproto[(size_t)bq * 4 * FF + i];
  for (int i = 4 * FF + threadIdx.x; i < 16 * FF; i += 128)
    protoS[i] = 0.0f;
  if (threadIdx.x < 4) p2S[threadIdx.x] = p2ws[bq * 4 + threadIdx.x];
  {
    int l = threadIdx.x;  // 128 threads == L rows
    const v4f* qr = (const v4f*)(qemb + ((size_t)bq * LL + l) * FF);
    float s = 0.0f;
    for (int f4 = 0; f4 < FF / 4; ++f4) {
      v4f v = qr[f4];
      s += v.x * v.x + v.y * v.y + v.z * v.z + v.w * v.w;
    }
    q2S[l] = s;
  }
  __syncthreads();

  int lane = threadIdx.x & 31;
  int wave = threadIdx.x >> 5;
  int sub  = lane & 15;               // M-row (A) / N-col (B) within tile
  int kh   = (lane >> 4) << 1;        // K half-select: 0 for lanes 0-15, 2 for 16-31

  for (int t = wave * 2; t < wave * 2 + 2; ++t) {
    int lrowA = t * 16 + sub;
    const float* arow = qemb + ((size_t)bq * LL + lrowA) * FF;
    const float* brow = protoS + sub * FF;   // rows >= 4 are zero
    v8f acc = {};
    for (int k0 = 0; k0 < FF; k0 += 4) {
      v2f a  = *(const v2f*)(arow + k0 + kh);   // global_load_b64
      v2f bf = *(const v2f*)(brow + k0 + kh);   // ds_load_b64 (8B aligned)
      acc = __builtin_amdgcn_wmma_f32_16x16x4_f32(
          /*neg_a=*/false, a, /*neg_b=*/false, bf,
          /*c_mod=*/(short)0, acc, /*reuse_a=*/false, /*reuse_b=*/false);
    }
    // C/D layout: VGPR r -> M = r (lanes 0-15) or M = 8+r (lanes 16-31); N = lane&15
    int n = sub;
    if (n < 4) {
      int mbase = (lane >= 16) ? 8 : 0;
      #pragma unroll
      for (int r = 0; r < 8; ++r) {
        int l = t * 16 + mbase + r;
        float qp = acc[r];
        tlog[((size_t)bq * LL + l) * 4 + n] = -(q2S[l] + p2S[n] - 2.0f * qp);
      }
    }
  }
}

// ------------- k5: BCE token loss, per-class argmax over L, span decode
__global__ void k5_loss_pred(const float* __restrict__ tlog, const int* __restrict__ q_ent,
                             const int* __restrict__ q_att, const int* __restrict__ sent_pred,
                             float* wsf, float* __restrict__ out) {
  int bq = blockIdx.x;
  int l  = threadIdx.x;   // 128 threads == L
  __shared__ float red[128];
  __shared__ float av[128];
  __shared__ int   ai[128];
  __shared__ int   idxs[4];

  float x[4];
  #pragma unroll
  for (int c = 0; c < 4; ++c) x[c] = tlog[((size_t)bq * LL + l) * 4 + c];
  float att = (float)q_att[bq * LL + l];
  int   ev  = q_ent[bq * LL + l];

  float bce = 0.0f;
  #pragma unroll
  for (int c = 0; c < 4; ++c) {
    float tgt = (ev == c + 1) ? 1.0f : 0.0f;
    float xv  = x[c];
    bce += fmaxf(xv, 0.0f) - xv * tgt + log1pf(__expf(-fabsf(xv)));
  }
  bce *= att;

  red[l] = bce; __syncthreads();
  for (int s = 64; s > 0; s >>= 1) { if (l < s) red[l] += red[l + s]; __syncthreads(); }
  float bce_tot = red[0]; __syncthreads();
  red[l] = att; __syncthreads();
  for (int s = 64; s > 0; s >>= 1) { if (l < s) red[l] += red[l + s]; __syncthreads(); }
  float att_tot = red[0]; __syncthreads();
  if (l == 0) {
    float per_ex = bce_tot / (att_tot * 4.0f + EPSF);
    atomicAdd(&wsf[1], per_ex * INV_BQ);
  }

  float pen = 1000.0f * (1.0f - att);
  for (int c = 0; c < 4; ++c) {
    av[l] = x[c] - pen; ai[l] = l; __syncthreads();
    for (int s = 64; s > 0; s >>= 1) {
      if (l < s) {
        float v1 = av[l], v2 = av[l + s];
        int   i1 = ai[l], i2 = ai[l + s];
        if (v2 > v1 || (v2 == v1 && i2 < i1)) { av[l] = v2; ai[l] = i2; }  // first-max ties
      }
      __syncthreads();
    }
    if (l == 0) idxs[c] = ai[0];
    __syncthreads();
  }

  int i0 = idxs[0], i1 = idxs[1], i2 = idxs[2], i3 = idxs[3];
  int p = 0;
  if (l == i0) p = 1;
  if (l > i0 && l <= i1) p = 2;
  if (l == i2) p = 3;
  if (l > i2 && l <= i3) p = 4;
  int fp = (p != 0) ? 4 * sent_pred[bq] + p : 0;
  out[1 + (size_t)bq * LL + l] = (float)fp;
}

// ---------------------------------------------------------------- k6: loss
__global__ void k6_final(const float* wsf, float* out) {
  out[0] = wsf[0] + wsf[1];
}

extern "C" void kernel_launch(void* const* d_in, const int* in_sizes, int n_in,
                              void* d_out, int out_size, void* d_ws, size_t ws_size,
                              hipStream_t stream) {
  (void)in_sizes; (void)n_in; (void)out_size; (void)ws_size;
  const float* sup   = (const float*)d_in[0];
  const float* qemb  = (const float*)d_in[1];
  const int*   s_ent = (const int*)d_in[2];
  const int*   q_ent = (const int*)d_in[3];
  const int*   s_att = (const int*)d_in[4];
  const int*   q_att = (const int*)d_in[5];
  const int*   label = (const int*)d_in[6];
  float* out = (float*)d_out;
  float* wsf = (float*)d_ws;
  int*   wsi = (int*)(wsf + OFF_END);   // sent_pred[B*NQ]

  hipLaunchKernelGGL(k0_init,        dim3(1),        dim3(32),  0, stream, wsf);
  hipLaunchKernelGGL(k1_sent_proto,  dim3(BB * NN),  dim3(256), 0, stream, sup, wsf + OFF_PROTO);
  hipLaunchKernelGGL(k2_sent,        dim3(BB * NQC), dim3(128), 0, stream, qemb, wsf + OFF_PROTO, label, wsf, wsi);
  hipLaunchKernelGGL(k3_token_proto, dim3(BB * NQC), dim3(256), 0, stream, sup, s_ent, s_att, label,
                     wsf + OFF_TPROTO, wsf + OFF_P2);
  hipLaunchKernelGGL(k4_qp_wmma,     dim3(BB * NQC), dim3(128), 0, stream, qemb, wsf + OFF_TPROTO,
                     wsf + OFF_P2, wsf + OFF_TLOGITS);
  hipLaunchKernelGGL(k5_loss_pred,   dim3(BB * NQC), dim3(128), 0, stream, wsf + OFF_TLOGITS,
                     q_ent, q_att, wsi, wsf, out);
  hipLaunchKernelGGL(k6_final,       dim3(1),        dim3(1),   0, stream, wsf, out);
}